// SmapeLoss1_34471407517908
// MI455X (gfx1250) — compile-verified
//
#include <hip/hip_runtime.h>

typedef __attribute__((ext_vector_type(2))) float v2f;
typedef __attribute__((ext_vector_type(4))) float v4f;
typedef __attribute__((ext_vector_type(8))) float v8f;

#define SMAPE_EPS 0.001f
#define NBLOCKS   4096
#define NTHREADS  256
#define NWAVES    (NTHREADS / 32)

// ---------------------------------------------------------------------------
// per-element SMAPE term: |t-p|/(|p|+|t|) if denom >= EPS else 0
// rcp is fine: masked-off lanes (den < EPS) are overwritten by the select.
// ---------------------------------------------------------------------------
__device__ __forceinline__ float smape_term(float p, float t) {
    float den = __builtin_fabsf(p) + __builtin_fabsf(t);
    float num = __builtin_fabsf(t - p);
    float r   = num * __builtin_amdgcn_rcpf(den);
    return (den >= SMAPE_EPS) ? r : 0.0f;
}

// ---------------------------------------------------------------------------
// Wave32 reduction of two per-lane partials using one V_WMMA_F32_16X16X4_F32.
// The wave's 64 partials form the 16x4 A matrix; B = ones(4x16) gives
// D[m][n] = sum_k A[m][k] (row sums, replicated across columns).
// C/D layout: VGPR v holds row v (lanes 0-15) / row v+8 (lanes 16-31),
// so sum(d[0..7]) in lane<16 covers rows 0-7, in lane>=16 rows 8-15.
// Must be called with EXEC all-ones (all 256 threads, no divergence).
// ---------------------------------------------------------------------------
__device__ __forceinline__ float wave_reduce_wmma(float ax, float ay) {
    v2f a; a.x = ax;   a.y = ay;
    v2f b; b.x = 1.0f; b.y = 1.0f;
    v8f c = {};
    c = __builtin_amdgcn_wmma_f32_16x16x4_f32(
            /*neg_a=*/false, a, /*neg_b=*/false, b,
            /*c_mod=*/(short)0, c, /*reuse_a=*/false, /*reuse_b=*/false);
    float s = ((c[0] + c[1]) + (c[2] + c[3])) + ((c[4] + c[5]) + (c[6] + c[7]));
    // rows 0-7 total lives in lanes 0-15, rows 8-15 total in lanes 16-31
    return __shfl(s, 0) + __shfl(s, 16);
}

// ---------------------------------------------------------------------------
// Kernel 1: streaming pass. Non-temporal float4 loads, grid-stride,
// per-block partial sum written to d_ws.
// ---------------------------------------------------------------------------
__global__ __launch_bounds__(NTHREADS)
void smape_partial_kernel(const float* __restrict__ pred,
                          const float* __restrict__ targ,
                          float* __restrict__ partials,
                          int nvec, int n) {
    const int tid    = blockIdx.x * NTHREADS + threadIdx.x;
    const int stride = gridDim.x * NTHREADS;

    float accx = 0.0f, accy = 0.0f;

    const v4f* __restrict__ p4 = (const v4f*)pred;
    const v4f* __restrict__ t4 = (const v4f*)targ;

    #pragma unroll 2
    for (int i = tid; i < nvec; i += stride) {
        v4f p = __builtin_nontemporal_load(p4 + i);   // global_load_b128, TH=NT
        v4f t = __builtin_nontemporal_load(t4 + i);
        accx += smape_term(p.x, t.x) + smape_term(p.z, t.z);
        accy += smape_term(p.y, t.y) + smape_term(p.w, t.w);
    }

    // scalar tail (n % 4 elements), handled by the first few global threads
    const int tail_base = nvec << 2;
    if (tid < n - tail_base) {
        accx += smape_term(pred[tail_base + tid], targ[tail_base + tid]);
    }

    // wave32 reduction via WMMA (all threads active -> EXEC all 1s)
    float w = wave_reduce_wmma(accx, accy);

    __shared__ float lds[NWAVES];
    const int wave = threadIdx.x >> 5;
    const int lane = threadIdx.x & 31;
    if (lane == 0) lds[wave] = w;
    __syncthreads();

    if (wave == 0) {
        float v = (lane < NWAVES) ? lds[lane] : 0.0f;
        v += __shfl_xor(v, 4);
        v += __shfl_xor(v, 2);
        v += __shfl_xor(v, 1);
        if (lane == 0) partials[blockIdx.x] = v;
    }
}

// ---------------------------------------------------------------------------
// Kernel 2: single block folds the 4096 block partials deterministically
// and applies the 200/n scale. Same WMMA wave reduction.
// ---------------------------------------------------------------------------
__global__ __launch_bounds__(NTHREADS)
void smape_final_kernel(const float* __restrict__ partials,
                        float* __restrict__ out,
                        int nparts, float scale) {
    float accx = 0.0f, accy = 0.0f;
    for (int i = threadIdx.x; i < nparts; i += 2 * NTHREADS) {
        accx += partials[i];
        int j = i + NTHREADS;
        if (j < nparts) accy += partials[j];
    }

    float w = wave_reduce_wmma(accx, accy);

    __shared__ float lds[NWAVES];
    const int wave = threadIdx.x >> 5;
    const int lane = threadIdx.x & 31;
    if (lane == 0) lds[wave] = w;
    __syncthreads();

    if (wave == 0) {
        float v = (lane < NWAVES) ? lds[lane] : 0.0f;
        v += __shfl_xor(v, 4);
        v += __shfl_xor(v, 2);
        v += __shfl_xor(v, 1);
        if (lane == 0) out[0] = v * scale;
    }
}

// ---------------------------------------------------------------------------
// launcher
// ---------------------------------------------------------------------------
extern "C" void kernel_launch(void* const* d_in, const int* in_sizes, int n_in,
                              void* d_out, int out_size, void* d_ws, size_t ws_size,
                              hipStream_t stream) {
    const float* pred = (const float*)d_in[0];
    const float* targ = (const float*)d_in[1];
    float* out        = (float*)d_out;
    float* partials   = (float*)d_ws;          // NBLOCKS floats = 16 KB

    const int n    = in_sizes[0];              // B*S elements
    const int nvec = n >> 2;                   // float4 count
    const float scale = 200.0f / (float)n;     // (200/S)/B == 200/n

    smape_partial_kernel<<<NBLOCKS, NTHREADS, 0, stream>>>(pred, targ, partials, nvec, n);
    smape_final_kernel<<<1, NTHREADS, 0, stream>>>(partials, out, NBLOCKS, scale);
}